// SGatedHebbianUnit_19172734009492
// MI455X (gfx1250) — compile-verified
//
#include <hip/hip_runtime.h>
#include <cmath>

typedef __attribute__((ext_vector_type(2))) float v2f;
typedef __attribute__((ext_vector_type(8))) float v8f;

constexpr int Ln = 4, Pn = 8, Bn = 256, Nn = 1024, Kn = 32;

__global__ void zero_f32(float* __restrict__ p, int n) {
    int i = blockIdx.x * 256 + threadIdx.x;
    if (i < n) p[i] = 0.0f;
}

// ---------------------------------------------------------------------------
// Kernel A: one block per (p,b).  Streams WR[p,b] (K x N) once:
//   - copies it into WR_new rows 0..K-1
//   - low[k]  = dot(WR[k,:], v_curr[src,b,:])
//   - lowh[k] = dot(WR[k,:], v_prev[src,b,:])
//   - writes dWR row (row K of WR_new) = m * xh / n_const
// ---------------------------------------------------------------------------
__global__ __launch_bounds__(256) void wr_pass(
    const float* __restrict__ WR, const float* __restrict__ v_prev,
    const float* __restrict__ v_curr, const int* __restrict__ pc,
    const int* __restrict__ srcA, float* __restrict__ WR_new,
    float* __restrict__ low_ws, float* __restrict__ lowh_ws, float n_const)
{
    int pb = blockIdx.x;
    int p = pb >> 8, b = pb & 255;
    int src = srcA[p];
    const float* wr  = WR + (size_t)pb * (Kn * Nn);
    const float* x   = v_curr + ((size_t)src * Bn + b) * Nn;
    const float* xh  = v_prev + ((size_t)src * Bn + b) * Nn;
    float* out = WR_new + (size_t)pb * ((Kn + 1) * Nn);

    __shared__ __align__(16) float xs[Nn];
    __shared__ __align__(16) float xhs[Nn];
    __shared__ float part[Kn][8];
    __shared__ float parth[Kn][8];

    int tid = threadIdx.x;
    for (int i = tid; i < Nn; i += 256) { xs[i] = x[i]; xhs[i] = xh[i]; }
    __syncthreads();

    int k = tid >> 3, j = tid & 7;                 // 8 threads per k-row
    const float4* row  = (const float4*)(wr + (size_t)k * Nn);
    float4*       orow = (float4*)(out + (size_t)k * Nn);
    const float4* xs4  = (const float4*)xs;
    const float4* xhs4 = (const float4*)xhs;

    float acc = 0.0f, acch = 0.0f;
    #pragma unroll 4
    for (int i = 0; i < 32; ++i) {
        int q = j + 8 * i;                         // float4 index within row
        float4 w = row[q];
        orow[q] = w;                               // fused copy to WR_new
        float4 xv = xs4[q], xhv = xhs4[q];
        acc  += w.x * xv.x  + w.y * xv.y  + w.z * xv.z  + w.w * xv.w;
        acch += w.x * xhv.x + w.y * xhv.y + w.z * xhv.z + w.w * xhv.w;
    }
    part[k][j] = acc; parth[k][j] = acch;
    __syncthreads();

    if (tid < Kn) {
        float s = 0.0f, sh = 0.0f;
        #pragma unroll
        for (int t = 0; t < 8; ++t) { s += part[tid][t]; sh += parth[tid][t]; }
        low_ws[(size_t)pb * Kn + tid]  = s;
        lowh_ws[(size_t)pb * Kn + tid] = sh;
    }

    // dWR row (row index K): m * xh / n_const
    float mf = (float)pc[b * Pn + p];
    float4 xv = xhs4[tid];
    float4 dv;
    dv.x = (mf * xv.x) / n_const; dv.y = (mf * xv.y) / n_const;
    dv.z = (mf * xv.z) / n_const; dv.w = (mf * xv.w) / n_const;
    ((float4*)(out + (size_t)Kn * Nn))[tid] = dv;
}

// ---------------------------------------------------------------------------
// Kernel B: one block (8 waves) per (p,b).
//   WMMA phase: inter[n] = WL[n,:]·low, pred[n] = WL[n,:]·lowh using
//   v_wmma_f32_16x16x4_f32.  A = broadcast(low chunk) (rows redundant),
//   B[k][j] = WL[n0+j][kc+k]  =>  D[:,j] = inter[n0+j]; lane l holds
//   inter[n0 + l%16] in every accumulator element.
//   B-operand loads for all 8 k-chunks are issued as a batch so the
//   compiler keeps 8 global_load_b64 in flight while WMMAs drain them.
//   Then: atomicAdd gates[p]*inter into mixed[dst,b,:], and write WL_new
//   copy + dWL column = m*(g*y - pred).
// ---------------------------------------------------------------------------
__global__ __launch_bounds__(256) void wl_pass(
    const float* __restrict__ WL, const float* __restrict__ v_curr,
    const float* __restrict__ gates, const int* __restrict__ pc,
    const int* __restrict__ dstA,
    const float* __restrict__ low_ws, const float* __restrict__ lowh_ws,
    float* __restrict__ WL_new, float* __restrict__ mixed, float g_const)
{
    int pb = blockIdx.x;
    int p = pb >> 8, b = pb & 255;
    int dst = dstA[p];
    const float* wl = WL + (size_t)pb * (Nn * Kn);
    float* out = WL_new + (size_t)pb * ((size_t)Nn * (Kn + 1));

    __shared__ float lows[Kn], lowhs[Kn];
    __shared__ float inter_s[Nn], pred_s[Nn], yhs[Nn];

    int tid = threadIdx.x;
    if (tid < Kn) {
        lows[tid]  = low_ws[(size_t)pb * Kn + tid];
        lowhs[tid] = lowh_ws[(size_t)pb * Kn + tid];
    }
    const float* yh = v_curr + ((size_t)dst * Bn + b) * Nn;
    for (int i = tid; i < Nn; i += 256) yhs[i] = yh[i];
    __syncthreads();

    int wave = tid >> 5, lane = tid & 31;
    int col = lane & 15, hi = lane >> 4;

    // A operands are loop-invariant: hoist all 8 chunk pairs into registers.
    v2f ai[8], ah[8];
    #pragma unroll
    for (int c8 = 0; c8 < 8; ++c8) {
        int kk = c8 * 4 + 2 * hi;
        ai[c8].x = lows[kk];  ai[c8].y = lows[kk + 1];
        ah[c8].x = lowhs[kk]; ah[c8].y = lowhs[kk + 1];
    }

    // 64 chunks of 16 n-rows; each wave does 8 chunks (uniform trip count).
    for (int c = wave; c < Nn / 16; c += 8) {
        int n0 = c * 16;
        const float* base = wl + (size_t)(n0 + col) * Kn + 2 * hi;

        // Batch-issue all 8 B-operand loads (independent) for pipelining.
        v2f bm[8];
        #pragma unroll
        for (int c8 = 0; c8 < 8; ++c8)
            bm[c8] = *(const v2f*)(base + c8 * 4);

        v8f ci = {0, 0, 0, 0, 0, 0, 0, 0};
        v8f cp = {0, 0, 0, 0, 0, 0, 0, 0};
        #pragma unroll
        for (int c8 = 0; c8 < 8; ++c8) {
            ci = __builtin_amdgcn_wmma_f32_16x16x4_f32(false, ai[c8], false, bm[c8],
                                                       (short)0, ci, false, false);
            cp = __builtin_amdgcn_wmma_f32_16x16x4_f32(false, ah[c8], false, bm[c8],
                                                       (short)0, cp, false, false);
        }
        if (hi == 0) {                // lanes 16..31 hold duplicates
            inter_s[n0 + col] = ci[0];
            pred_s[n0 + col]  = cp[0];
        }
    }
    __syncthreads();

    // mixed[dst,b,:] += gates[p] * inter  (exactly 2 contributors per slot:
    // float add is commutative so the result is bitwise deterministic)
    float gp = gates[p];
    float* mrow = mixed + ((size_t)dst * Bn + b) * Nn;
    for (int i = tid; i < Nn; i += 256) atomicAdd(&mrow[i], gp * inter_s[i]);

    // WL_new: copy 32 cols + dWL column.  One wave per row: 128B coalesced.
    float mf = (float)pc[b * Pn + p];
    for (int r = wave; r < Nn; r += 8) {
        float w = wl[(size_t)r * Kn + lane];
        out[(size_t)r * (Kn + 1) + lane] = w;
        if (lane == 0)
            out[(size_t)r * (Kn + 1) + Kn] = mf * (g_const * yhs[r] - pred_s[r]);
    }
}

__global__ void tanh_k(const float* __restrict__ mixed, float* __restrict__ v_next, int n) {
    int i = blockIdx.x * 256 + threadIdx.x;
    if (i < n) v_next[i] = tanhf(mixed[i]);
}

extern "C" void kernel_launch(void* const* d_in, const int* in_sizes, int n_in,
                              void* d_out, int out_size, void* d_ws, size_t ws_size,
                              hipStream_t stream) {
    const float* WL     = (const float*)d_in[0];
    const float* WR     = (const float*)d_in[1];
    const float* v_prev = (const float*)d_in[2];
    const float* v_curr = (const float*)d_in[3];
    const float* gates  = (const float*)d_in[4];
    const int*   pc     = (const int*)d_in[5];
    const int*   dstA   = (const int*)d_in[6];
    const int*   srcA   = (const int*)d_in[7];

    float* outp   = (float*)d_out;
    float* v_next = outp;                                          // [L,B,N]
    float* WL_new = outp + (size_t)Ln * Bn * Nn;                   // [P,B,N,K+1]
    float* WR_new = WL_new + (size_t)Pn * Bn * Nn * (Kn + 1);      // [P,B,K+1,N]

    float* ws      = (float*)d_ws;
    float* low_ws  = ws;                                           // P*B*K
    float* lowh_ws = low_ws + (size_t)Pn * Bn * Kn;                // P*B*K
    float* mixed   = lowh_ws + (size_t)Pn * Bn * Kn;               // L*B*N

    const double r = 0.999;
    float g_const = (float)(0.5 * (log1p(r) - log1p(-r)) / r);
    float n_const = (float)((double)Nn * r * r);

    int lbn = Ln * Bn * Nn;
    zero_f32<<<lbn / 256, 256, 0, stream>>>(mixed, lbn);
    wr_pass<<<Pn * Bn, 256, 0, stream>>>(WR, v_prev, v_curr, pc, srcA,
                                         WR_new, low_ws, lowh_ws, n_const);
    wl_pass<<<Pn * Bn, 256, 0, stream>>>(WL, v_curr, gates, pc, dstA,
                                         low_ws, lowh_ws, WL_new, mixed, g_const);
    tanh_k<<<lbn / 256, 256, 0, stream>>>(mixed, v_next, lbn);
}